// RNNTaggerModel_38732015075628
// MI455X (gfx1250) — compile-verified
//
#include <hip/hip_runtime.h>

#define NEG_INF (-1e4f)
#define BB 64
#define TT 512
#define LL 512

typedef __attribute__((ext_vector_type(2))) float v2f;
typedef __attribute__((ext_vector_type(8))) float v8f;

// ---------------------------------------------------------------------------
// Workspace layout (bytes)
//   [0,4)          grid-barrier counter
//   [1024, +256KB) alphas double buffer AB[2][64][512] f32
//   [oE, +1MB)     Esw[f/2][to] float2 = {exp(trans[to][f]), exp(trans[to][f+1])}
//   [oLAST, +256)  last[64] int  (viterbi terminal argmax)
//   [oBP, +32MB)   bp[t][b][to] u16 backpointers
// ---------------------------------------------------------------------------
#define O_AB   1024
#define O_E    (O_AB + 2 * BB * LL * 4)
#define O_LAST (O_E + LL * LL * 4)
#define O_BP   (O_LAST + 1024)

// ---------------- prep: exp+swizzle transitions, init alphas, zero barrier ----
__global__ __launch_bounds__(256)
void crf_prep(const float* __restrict__ trans, float2* __restrict__ Esw,
              float* __restrict__ AB0, unsigned* __restrict__ cnt,
              const int* __restrict__ start_p) {
  int idx = blockIdx.x * 256 + threadIdx.x;
  if (idx == 0) *cnt = 0u;
  if (idx < (LL / 2) * LL) {
    int f2 = idx >> 9;          // pair index along "from"
    int to = idx & (LL - 1);
    float2 e;
    e.x = __expf(trans[to * LL + 2 * f2]);
    e.y = __expf(trans[to * LL + 2 * f2 + 1]);
    Esw[(size_t)f2 * LL + to] = e;
  }
  if (idx < BB * LL) {
    int l = idx & (LL - 1);
    AB0[idx] = (l == *start_p) ? 0.f : NEG_INF;
  }
}

// ---------------- forward: persistent exp-domain GEMM scan (WMMA f32) --------
// 32 WGs x 128 threads. WG g: batch group bg=g>>3 (16 rows), col slice cs=g&7
// (64 cols). 4 waves, each owns one 16x16 output tile; K=512 -> 128 wmma/step.
__global__ __launch_bounds__(128)
void crf_forward_wmma(const float* __restrict__ unary,
                      const float2* __restrict__ Esw,
                      const int* __restrict__ lengths,
                      float* __restrict__ AB, unsigned* __restrict__ cnt) {
  __shared__ float p_lds[16 * LL];     // 32KB: p rows for this batch group
  __shared__ float pmax[16][8];
  __shared__ float mrow[16];
  __shared__ int len_s[16];

  const int tid = threadIdx.x;
  const int g = blockIdx.x;
  const int bg = g >> 3;               // 0..3
  const int cs = g & 7;                // 0..7
  const int lane = tid & 31;
  const int wave = tid >> 5;
  const int ln15 = lane & 15;
  const int khalf = lane >> 4;         // 0|1 -> K sub-pair (ISA 16x16x4 layout)
  const int wcol = cs * 64 + wave * 16;  // global column base of this wave's tile
  const unsigned nwg = 32;

  if (tid < 16) len_s[tid] = lengths[bg * 16 + tid];
  __syncthreads();

  for (int t = 0; t < TT; ++t) {
    const float* Acur = AB + (size_t)(t & 1) * BB * LL;
    float* Anext = AB + (size_t)((t + 1) & 1) * BB * LL;

    // Phase A: per-batch-row max (stability shift; any per-row shift is exact)
    {
      int row = tid >> 3, sub = tid & 7;
      const float* ar = Acur + (bg * 16 + row) * LL;
      float m = -3.4e38f;
      for (int j = sub; j < LL; j += 8) m = fmaxf(m, ar[j]);
      pmax[row][sub] = m;
    }
    __syncthreads();
    if (tid < 16) {
      float m = pmax[tid][0];
#pragma unroll
      for (int s = 1; s < 8; ++s) m = fmaxf(m, pmax[tid][s]);
      mrow[tid] = m;
    }
    __syncthreads();

    // Phase B: p = exp(alpha - m) into LDS
    for (int f = tid; f < 16 * LL; f += 128) {
      int row = f >> 9, col = f & (LL - 1);
      p_lds[f] = __expf(Acur[(bg * 16 + row) * LL + col] - mrow[row]);
    }
    __syncthreads();

    // Phase C: 16x16 tile, K=512 via v_wmma_f32_16x16x4_f32.
    // A (16x4 f32): lane ln15 = M, khalf selects K-pair -> float2 from LDS.
    // B (4x16 f32): lane ln15 = N, khalf selects K-pair -> float2 from Esw.
    v8f acc = (v8f)0.0f;
    const float* pa = p_lds + ln15 * LL + khalf * 2;
    const float2* pb = Esw + (size_t)khalf * LL + (wcol + ln15);
#pragma unroll 4
    for (int k = 0; k < LL; k += 4) {
      float2 av = *(const float2*)(pa + k);
      float2 bv = pb[(size_t)(k >> 1) * LL];
      v2f a; a[0] = av.x; a[1] = av.y;
      v2f b; b[0] = bv.x; b[1] = bv.y;
      acc = __builtin_amdgcn_wmma_f32_16x16x4_f32(
          /*neg_a=*/false, a, /*neg_b=*/false, b,
          /*c_mod=*/(short)0, acc, /*reuse_a=*/false, /*reuse_b=*/false);
    }

    // Epilogue: alpha' = m + log(S) + unary; freeze rows past their length.
    // C/D layout: vgpr r -> row r (+8 for lanes 16-31), col = ln15.
    {
      int rowbase = khalf * 8;
      int col = wcol + ln15;
#pragma unroll
      for (int r = 0; r < 8; ++r) {
        int row = rowbase + r;
        int b_ = bg * 16 + row;
        float S = acc[r];
        float na = mrow[row] + __logf(fmaxf(S, 1e-37f)) +
                   unary[((size_t)b_ * TT + t) * LL + col];
        float old = Acur[b_ * LL + col];
        Anext[b_ * LL + col] = (t < len_s[row]) ? na : old;
      }
    }

    // Grid barrier (monotone counter; WS counter zeroed by prep each call)
    __threadfence();
    __syncthreads();
    if (tid == 0) {
      atomicAdd(cnt, 1u);
      unsigned target = nwg * (unsigned)(t + 1);
      while (__hip_atomic_load(cnt, __ATOMIC_ACQUIRE, __HIP_MEMORY_SCOPE_AGENT) <
             target) {
        __builtin_amdgcn_s_sleep(2);
      }
      __threadfence();
    }
    __syncthreads();
  }
}

// ---------------- terminal logZ = logsumexp(alpha_final + trans[end,:]) ------
__global__ __launch_bounds__(256)
void crf_logz(const float* __restrict__ AB0, const float* __restrict__ trans,
              float* __restrict__ out_logZ, const int* __restrict__ end_p) {
  __shared__ float red[256];
  int b = blockIdx.x, tid = threadIdx.x;
  const float* a = AB0 + b * LL;
  const float* tr = trans + (*end_p) * LL;
  float m = -3.4e38f;
  for (int l = tid; l < LL; l += 256) m = fmaxf(m, a[l] + tr[l]);
  red[tid] = m;
  __syncthreads();
  for (int s = 128; s > 0; s >>= 1) {
    if (tid < s) red[tid] = fmaxf(red[tid], red[tid + s]);
    __syncthreads();
  }
  m = red[0];
  __syncthreads();
  float s = 0.f;
  for (int l = tid; l < LL; l += 256) s += __expf(a[l] + tr[l] - m);
  red[tid] = s;
  __syncthreads();
  for (int st = 128; st > 0; st >>= 1) {
    if (tid < st) red[tid] += red[tid + st];
    __syncthreads();
  }
  if (tid == 0) out_logZ[b] = m + __logf(red[0]);
}

// ---------------- gold path score -------------------------------------------
__global__ __launch_bounds__(128)
void crf_gold(const float* __restrict__ unary, const float* __restrict__ trans,
              const int* __restrict__ lengths, const int* __restrict__ tags,
              float* __restrict__ out_gold, const int* __restrict__ start_p,
              const int* __restrict__ end_p) {
  __shared__ float red[128];
  int b = blockIdx.x, tid = threadIdx.x;
  int len = lengths[b];
  int start_idx = *start_p;
  float s = 0.f;
  for (int t = tid; t < len; t += 128) {
    int cur = tags[b * TT + t];
    int prev = (t == 0) ? start_idx : tags[b * TT + t - 1];
    s += trans[cur * LL + prev] + unary[((size_t)b * TT + t) * LL + cur];
  }
  red[tid] = s;
  __syncthreads();
  for (int st = 64; st > 0; st >>= 1) {
    if (tid < st) red[tid] += red[tid + st];
    __syncthreads();
  }
  if (tid == 0)
    out_gold[b] = red[0] + trans[(*end_p) * LL + tags[b * TT + len - 1]];
}

// ---------------- viterbi scan (max-plus, LDS-staged trans) ------------------
// One WG per batch, 256 threads, 2 "to" states per thread. trans streamed from
// L2 into LDS in 512x64 chunks, row stride 66 floats (bank-conflict-free b64).
__global__ __launch_bounds__(256)
void crf_viterbi(const float* __restrict__ unary, const float* __restrict__ trans,
                 const int* __restrict__ lengths, unsigned short* __restrict__ bp,
                 int* __restrict__ last, float* __restrict__ pscore,
                 const int* __restrict__ start_p, const int* __restrict__ end_p) {
  extern __shared__ float tl[];        // [512][66] dynamic, 135168 B
  __shared__ float va_s[LL];
  __shared__ float rv[256];
  __shared__ int ri[256];

  int b = blockIdx.x, tid = threadIdx.x;
  int len = lengths[b];
  int start_idx = *start_p, end_idx = *end_p;
  const int to0 = tid, to1 = tid + 256;

  for (int l = tid; l < LL; l += 256) va_s[l] = (l == start_idx) ? 0.f : NEG_INF;
  __syncthreads();

  for (int t = 0; t < TT; ++t) {
    float best0 = -3.4e38f, best1 = -3.4e38f;
    int arg0 = 0, arg1 = 0;
    for (int cfo = 0; cfo < LL; cfo += 64) {
      __syncthreads();  // tl reuse
      // stage trans[:, cfo..cfo+63] into LDS (coalesced b128 global reads)
      for (int idx = tid; idx < LL * 16; idx += 256) {
        int to = idx >> 4, q = idx & 15;
        float4 v = *(const float4*)(trans + to * LL + cfo + q * 4);
        float2* dst = (float2*)(tl + to * 66 + q * 4);
        dst[0] = make_float2(v.x, v.y);
        dst[1] = make_float2(v.z, v.w);
      }
      __syncthreads();
#pragma unroll 8
      for (int fo = 0; fo < 64; fo += 2) {
        float2 va2 = *(const float2*)(va_s + cfo + fo);
        float2 ta = *(const float2*)(tl + to0 * 66 + fo);
        float2 tb = *(const float2*)(tl + to1 * 66 + fo);
        float s;
        s = va2.x + ta.x; if (s > best0) { best0 = s; arg0 = cfo + fo; }
        s = va2.y + ta.y; if (s > best0) { best0 = s; arg0 = cfo + fo + 1; }
        s = va2.x + tb.x; if (s > best1) { best1 = s; arg1 = cfo + fo; }
        s = va2.y + tb.y; if (s > best1) { best1 = s; arg1 = cfo + fo + 1; }
      }
    }
    __syncthreads();
    bool active = (t < len);
    if (active) {
      va_s[to0] = best0 + unary[((size_t)b * TT + t) * LL + to0];
      va_s[to1] = best1 + unary[((size_t)b * TT + t) * LL + to1];
    }
    size_t bprow = ((size_t)t * BB + b) * LL;
    bp[bprow + to0] = (unsigned short)(active ? arg0 : to0);
    bp[bprow + to1] = (unsigned short)(active ? arg1 : to1);
    __syncthreads();
  }

  // terminal max/argmax (first-occurrence tie-break)
  float v0 = va_s[to0] + trans[end_idx * LL + to0];
  float v1 = va_s[to1] + trans[end_idx * LL + to1];
  float bv = v0; int bi = to0;
  if (v1 > bv) { bv = v1; bi = to1; }
  rv[tid] = bv; ri[tid] = bi;
  __syncthreads();
  for (int s = 128; s > 0; s >>= 1) {
    if (tid < s) {
      bool take = (rv[tid + s] > rv[tid]) ||
                  (rv[tid + s] == rv[tid] && ri[tid + s] < ri[tid]);
      if (take) { rv[tid] = rv[tid + s]; ri[tid] = ri[tid + s]; }
    }
    __syncthreads();
  }
  if (tid == 0) { last[b] = ri[0]; pscore[b] = rv[0]; }
}

// ---------------- backtrack: LDS-staged pointer chase ------------------------
__global__ __launch_bounds__(256)
void crf_backtrack(const unsigned short* __restrict__ bp,
                   const int* __restrict__ last, float* __restrict__ out_path) {
  __shared__ unsigned short chunk[32 * LL];  // 32KB
  __shared__ int carry_s;
  int b = blockIdx.x, tid = threadIdx.x;
  if (tid == 0) carry_s = last[b];
  for (int c = 15; c >= 0; --c) {
    int t0 = c * 32;
    __syncthreads();
    for (int idx = tid; idx < 32 * (LL / 2); idx += 256) {
      int tt = idx / (LL / 2), j = idx % (LL / 2);
      ((unsigned int*)chunk)[tt * (LL / 2) + j] =
          ((const unsigned int*)(bp + ((size_t)(t0 + tt) * BB + b) * LL))[j];
    }
    __syncthreads();
    if (tid == 0) {
      int carry = carry_s;
      for (int tt = 31; tt >= 0; --tt) {
        out_path[b * TT + t0 + tt] = (float)carry;   // emit, then step back
        carry = chunk[tt * LL + carry];
      }
      carry_s = carry;
    }
  }
}

// ---------------------------------------------------------------------------
extern "C" void kernel_launch(void* const* d_in, const int* in_sizes, int n_in,
                              void* d_out, int out_size, void* d_ws,
                              size_t ws_size, hipStream_t stream) {
  (void)in_sizes; (void)n_in; (void)out_size; (void)ws_size;
  const float* unary  = (const float*)d_in[0];
  const float* trans  = (const float*)d_in[1];
  const int* lengths  = (const int*)d_in[2];
  const int* tags     = (const int*)d_in[3];
  const int* start_p  = (const int*)d_in[4];
  const int* end_p    = (const int*)d_in[5];

  float* out        = (float*)d_out;
  float* out_logZ   = out;                       // [64]
  float* out_gold   = out + BB;                  // [64]
  float* out_path   = out + 2 * BB;              // [64*512] as float
  float* out_pscore = out + 2 * BB + BB * TT;    // [64]

  char* ws = (char*)d_ws;
  unsigned* cnt          = (unsigned*)(ws);
  float* AB              = (float*)(ws + O_AB);
  float2* Esw            = (float2*)(ws + O_E);
  int* last              = (int*)(ws + O_LAST);
  unsigned short* bp     = (unsigned short*)(ws + O_BP);

  crf_prep<<<512, 256, 0, stream>>>(trans, Esw, AB, cnt, start_p);
  crf_forward_wmma<<<32, 128, 0, stream>>>(unary, Esw, lengths, AB, cnt);
  crf_logz<<<64, 256, 0, stream>>>(AB, trans, out_logZ, end_p);
  crf_gold<<<64, 128, 0, stream>>>(unary, trans, lengths, tags, out_gold,
                                   start_p, end_p);
  crf_viterbi<<<64, 256, 512 * 66 * 4, stream>>>(unary, trans, lengths, bp, last,
                                                 out_pscore, start_p, end_p);
  crf_backtrack<<<64, 256, 0, stream>>>(bp, last, out_path);
}